// Retinal_NET_43181601194477
// MI455X (gfx1250) — compile-verified
//
#include <hip/hip_runtime.h>
#include <math.h>

// ---------------------------------------------------------------------------
// Retinal_NET pipeline for MI455X (gfx1250, wave32).
//
// Stage 1 (bandwidth-dominant): 5x5 stride-5 block conv on x[32,200,100,100].
//   256 MB streamed once (non-temporal b128 loads) into LDS -> ~11us HBM floor.
// Stage 2 (WMMA): causal 31-tap temporal conv == banded-Toeplitz matmul
//   first = Toep(w)[200x200] @ s[200x400] per batch, fp32 via
//   V_WMMA_F32_16X16X4_F32, 12 K-steps of 4 per 16x16 tile, fully unrolled,
//   branch-free predication so EXEC stays all-ones and loads pipeline.
// Stage 3: 4x4 block conv -> flat[.,.,25], two 12-channel 31-tap causal
//   convs, sigmoid/abs combine, row/col dot products -> out[32,200,2].
// ---------------------------------------------------------------------------

typedef float v2f __attribute__((ext_vector_type(2)));
typedef float v4f __attribute__((ext_vector_type(4)));
typedef float v8f __attribute__((ext_vector_type(8)));

#define B_  32
#define T_  200
#define HW_ 100
#define S_LOC 400   // 20*20
#define F_LOC 25    // 5*5
#define CH_  12

// ------------------------------ Kernel 1 -----------------------------------
// One workgroup per (b,t) plane: stage 100x100 fp32 into LDS, emit 20x20.
__global__ void k1_block5(const float* __restrict__ x,
                          const float* __restrict__ space_w,
                          float* __restrict__ s_out) {
  __shared__ float tile[HW_ * HW_];           // 40 KB
  const int bt  = blockIdx.x;                 // 0..6399
  const int tid = threadIdx.x;                // 0..255

  // Coalesced staged load: 10000 floats = 2500 x b128 (plane is 16B aligned).
  // Non-temporal: x is streamed exactly once and exceeds L2 capacity.
  const v4f* src4 = (const v4f*)(x + (size_t)bt * (HW_ * HW_));
  v4f* dst4 = (v4f*)tile;
  for (int i = tid; i < (HW_ * HW_) / 4; i += 256)
    dst4[i] = __builtin_nontemporal_load(&src4[i]);

  float sw[25];
#pragma unroll
  for (int i = 0; i < 25; ++i) sw[i] = space_w[i];

  __syncthreads();

  for (int o = tid; o < S_LOC; o += 256) {
    const int p = o / 20, q = o % 20;
    float acc = 0.f;
#pragma unroll
    for (int a = 0; a < 5; ++a) {
      const float* row = &tile[(5 * p + a) * HW_ + 5 * q];
#pragma unroll
      for (int c = 0; c < 5; ++c) acc += row[c] * sw[a * 5 + c];
    }
    s_out[(size_t)bt * S_LOC + o] = acc;
  }
}

// ------------------------------ Kernel 2 -----------------------------------
// first[b,t,n] = relu( sum_j w[30-(t-j)] * s[b,j,n] ),  0 <= t-j <= 30.
// One wave per 16x16 output tile. A = Toeplitz(temporal_w) pulled from a
// 32-entry LDS array whose slot 31 is 0.0 (out-of-band entries map there via
// unsigned compare). B rows are clamped and zero-selected -- no divergence,
// so EXEC is all-ones at every WMMA and loads can batch across K-steps.
__global__ void k2_temporal_wmma(const float* __restrict__ s_in,
                                 const float* __restrict__ tw,
                                 float* __restrict__ first) {
  __shared__ float twl[32];
  const int nt = blockIdx.x;          // 0..24   (N tile: 16 spatial locs)
  const int mt = blockIdx.y;          // 0..12   (M tile: 16 time steps)
  const int b  = blockIdx.z;          // 0..31
  const int t0 = mt * 16;
  const int n0 = nt * 16;

  const int l    = threadIdx.x;       // 0..31, single full wave
  const int m    = l & 15;
  const int half = l >> 4;            // lanes 16-31: K+2 (A/B), M+8 (D)
  const int kA   = half * 2;

  twl[l] = (l < 31) ? tw[l] : 0.f;    // single wave: DScnt ordering suffices
  __syncthreads();

  const float* sB = s_in + (size_t)b * T_ * S_LOC;
  const int n = n0 + m;

  v8f c = {0.f, 0.f, 0.f, 0.f, 0.f, 0.f, 0.f, 0.f};

#pragma unroll
  for (int kt = 0; kt < 12; ++kt) {
    const int j0 = t0 - 32 + kt * 4;

    // A tile [16x4]: A[m,k] = tw[30-d], d = (t0+m)-(j0+k).
    // 30-d = 4*kt + kA - m - 2 ; out-of-range (incl. negative) -> slot 31.
    const unsigned i0 = (unsigned)(4 * kt + kA - m - 2);
    const unsigned i1 = i0 + 1u;      // note: i0==0xFFFFFFFF -> i1==0 (valid)
    v2f a;
    a.x = twl[i0 > 30u ? 31u : i0];
    a.y = twl[i1 > 30u ? 31u : i1];

    // B tile [4x16]: B[k,n] = s[b, j0+k, n0+n]; rows outside [0,200) -> 0.
    const int jb  = j0 + kA;
    const unsigned u0 = (unsigned)jb, u1 = (unsigned)(jb + 1);
    const int jc0 = (u0 < (unsigned)T_) ? jb : 0;
    const int jc1 = (u1 < (unsigned)T_) ? (jb + 1) : 0;
    const float v0 = sB[(size_t)jc0 * S_LOC + n];
    const float v1 = sB[(size_t)jc1 * S_LOC + n];
    v2f bb;
    bb.x = (u0 < (unsigned)T_) ? v0 : 0.f;
    bb.y = (u1 < (unsigned)T_) ? v1 : 0.f;

    c = __builtin_amdgcn_wmma_f32_16x16x4_f32(
        /*neg_a=*/false, a, /*neg_b=*/false, bb,
        /*c_mod=*/(short)0, c, /*reuse_a=*/false, /*reuse_b=*/false);
  }

  // D layout: VGPR v -> M = v + 8*half, N = lane&15.
  float* dst = first + (size_t)b * T_ * S_LOC;
#pragma unroll
  for (int v = 0; v < 8; ++v) {
    const int t = t0 + v + half * 8;
    if (t < T_) dst[(size_t)t * S_LOC + n] = fmaxf(c[v], 0.f);
  }
}

// ------------------------------ Kernel 3 -----------------------------------
__device__ __forceinline__ float sigmoidf(float v) {
  return 1.f / (1.f + expf(-v));
}

__global__ void k3_tail(const float* __restrict__ first,
                        const float* __restrict__ last_w,
                        const float* __restrict__ ama_create_w,
                        const float* __restrict__ gang_create_w,
                        const float* __restrict__ ama_kw,
                        const float* __restrict__ ama_kb,
                        const float* __restrict__ ama_alpha_w,
                        const float* __restrict__ gang_kw,
                        const float* __restrict__ gang_kb,
                        const float* __restrict__ row_w,
                        const float* __restrict__ col_w,
                        float* __restrict__ out) {
  __shared__ float flat[T_ * F_LOC];   // 20.0 KB
  __shared__ float tot[T_ * CH_];      //  9.6 KB
  const int b   = blockIdx.x;
  const int tid = threadIdx.x;

  float lw[16];
#pragma unroll
  for (int i = 0; i < 16; ++i) lw[i] = last_w[i];

  const float* fb = first + (size_t)b * T_ * S_LOC;

  // Stage A: 4x4 block conv (20x20 -> 5x5) per time step.
  for (int idx = tid; idx < T_ * F_LOC; idx += 256) {
    const int t = idx / F_LOC, o = idx % F_LOC;
    const int pp = o / 5, qq = o % 5;
    const float* base = fb + (size_t)t * S_LOC + (4 * pp) * 20 + 4 * qq;
    float acc = 0.f;
#pragma unroll
    for (int a = 0; a < 4; ++a)
#pragma unroll
      for (int cc = 0; cc < 4; ++cc) acc += base[a * 20 + cc] * lw[a * 4 + cc];
    flat[idx] = acc;
  }
  __syncthreads();

  // Stage B: causal convs on ama_pre = flat[2m]*acw and gang_pre =
  // flat[2m+1]*gcw (roll(-1)), then tot = sigmoid(gang - |ama|).
  const float acw = ama_create_w[0], gcw = gang_create_w[0];
  const float ab = ama_kb[0], gb = gang_kb[0], alpha = ama_alpha_w[0];
  for (int idx = tid; idx < T_ * CH_; idx += 256) {
    const int t = idx / CH_, m = idx % CH_;
    float ac = ab, gc = gb;
#pragma unroll 1
    for (int d = 0; d < 31; ++d) {        // tap d weights sample at t-30+d
      const int tau = t - 30 + d;
      const int tc  = tau < 0 ? 0 : tau;  // branch-free: clamp + zero-select
      const float f0 = flat[tc * F_LOC + 2 * m];
      const float f1 = flat[tc * F_LOC + 2 * m + 1];
      const float gate = (tau >= 0) ? 1.f : 0.f;
      ac += ama_kw[d]  * (f0 * acw) * gate;
      gc += gang_kw[d] * (f1 * gcw) * gate;
    }
    const float ama = alpha * sigmoidf(ac);
    tot[idx] = sigmoidf(gc - fabsf(ama));
  }
  __syncthreads();

  // Stage C: rows/cols dot products -> out[b,t,{0,1}].
  for (int t = tid; t < T_; t += 256) {
    float r = 0.f, csum = 0.f;
#pragma unroll
    for (int m = 0; m < CH_; ++m) {
      const float v = tot[t * CH_ + m];
      r += v * row_w[m];
      csum += v * col_w[m];
    }
    out[((size_t)b * T_ + t) * 2 + 0] = r;
    out[((size_t)b * T_ + t) * 2 + 1] = csum;
  }
}

// ------------------------------ Launch -------------------------------------
extern "C" void kernel_launch(void* const* d_in, const int* in_sizes, int n_in,
                              void* d_out, int out_size, void* d_ws, size_t ws_size,
                              hipStream_t stream) {
  const float* x        = (const float*)d_in[0];
  const float* space_w  = (const float*)d_in[1];
  const float* temporal = (const float*)d_in[2];
  const float* last_w   = (const float*)d_in[3];
  const float* ama_cw   = (const float*)d_in[4];
  const float* gang_cw  = (const float*)d_in[5];
  const float* ama_kw   = (const float*)d_in[6];
  const float* ama_kb   = (const float*)d_in[7];
  const float* ama_aw   = (const float*)d_in[8];
  const float* gang_kw  = (const float*)d_in[9];
  const float* gang_kb  = (const float*)d_in[10];
  const float* row_w    = (const float*)d_in[11];
  const float* col_w    = (const float*)d_in[12];
  float* out = (float*)d_out;

  const size_t nS = (size_t)B_ * T_ * S_LOC;           // 2.56M floats
  if (ws_size < 2 * nS * sizeof(float)) return;        // need ~20.5 MB scratch
  float* s_buf = (float*)d_ws;
  float* first = s_buf + nS;

  // K1: 6400 planes, 256 threads each.
  k1_block5<<<dim3(B_ * T_), dim3(256), 0, stream>>>(x, space_w, s_buf);

  // K2: one wave per 16x16 tile: 25 N-tiles x 13 M-tiles x 32 batches.
  k2_temporal_wmma<<<dim3(25, 13, B_), dim3(32), 0, stream>>>(s_buf, temporal, first);

  // K3: one workgroup per batch.
  k3_tail<<<dim3(B_), dim3(256), 0, stream>>>(first, last_w, ama_cw, gang_cw,
                                              ama_kw, ama_kb, ama_aw,
                                              gang_kw, gang_kb, row_w, col_w, out);
}